// SlaterPooling_83227876262134
// MI455X (gfx1250) — compile-verified
//
#include <hip/hip_runtime.h>
#include <hip/hip_bf16.h>

// CDNA5 / gfx1250 Slater determinant kernel.
// One wave32 per (batch, conf): gather 32x32 matrix into LDS, blocked LU
// with partial pivoting; the 16x16x16 Schur update runs on the matrix core
// via V_WMMA_F32_16X16X4_F32 (4 chained K=4 steps).

typedef __attribute__((ext_vector_type(2))) float v2f;
typedef __attribute__((ext_vector_type(8))) float v8f;

#define NB    1024
#define NCONF 128
#define NMO   96
#define NELEC 64
#define N     32
#define NH    16
#define LDSP  33   // padded LDS row stride (bank-conflict free column walks)

__global__ __launch_bounds__(32)
void slater_det_kernel(const float* __restrict__ mo,
                       const int*   __restrict__ cfg_up,
                       const int*   __restrict__ cfg_dn,
                       float*       __restrict__ out) {
  __shared__ float As[N * LDSP];
  const int lane = threadIdx.x;
  const int bc   = blockIdx.x;
  const int b    = bc / NCONF;
  const int c    = bc % NCONF;

  float result = 1.0f;

  for (int spin = 0; spin < 2; ++spin) {
    const int* cfg = (spin == 0 ? cfg_up : cfg_dn) + c * N;
    const int  col = cfg[lane];                         // lane j -> column j
    const float* mobase = mo + ((size_t)b * NELEC + (size_t)spin * N) * NMO;

    // ---- gather: As[i][lane] = mo[b, spin*32 + i, cfg[c][lane]] ----
    #pragma unroll 4
    for (int i = 0; i < N; ++i)
      As[i * LDSP + lane] = mobase[i * NMO + col];
    __syncthreads();

    float det = 1.0f;

    // ---- panel LU with partial pivoting, columns 0..15, rows k..31 ----
    for (int k = 0; k < NH; ++k) {
      // argmax |As[i][k]| over i = k..31 (butterfly, commutative tie-break)
      float v = -1.0f; int p = 1 << 20;
      if (lane >= k) { v = fabsf(As[lane * LDSP + k]); p = lane; }
      #pragma unroll
      for (int off = 16; off > 0; off >>= 1) {
        float ov = __shfl_xor(v, off);
        int   op = __shfl_xor(p, off);
        if (ov > v || (ov == v && op < p)) { v = ov; p = op; }
      }
      // full-row swap k <-> p (lane handles one column); benign if p == k
      {
        float tk = As[k * LDSP + lane];
        float tp = As[p * LDSP + lane];
        As[k * LDSP + lane] = tp;
        As[p * LDSP + lane] = tk;
      }
      __syncthreads();
      float piv = As[k * LDSP + k];
      det *= piv;
      if (p != k) det = -det;
      float inv = (piv != 0.0f) ? 1.0f / piv : 0.0f;   // singular -> det = 0
      float lik = 0.0f;
      if (lane > k) {
        lik = As[lane * LDSP + k] * inv;
        As[lane * LDSP + k] = lik;
      }
      __syncthreads();
      if (lane > k)
        for (int j = k + 1; j < NH; ++j)
          As[lane * LDSP + j] -= lik * As[k * LDSP + j];
      __syncthreads();
    }

    // ---- trsm: A12 (rows 0..15, cols 16..31) = L11^{-1} * A12 ----
    {
      const int j    = 16 + (lane & 15);
      const int half = lane >> 4;       // halves split rows by parity
      for (int k = 0; k < NH; ++k) {
        float akj = As[k * LDSP + j];
        for (int i = k + 1 + half; i < NH; i += 2)
          As[i * LDSP + j] -= As[i * LDSP + k] * akj;
        __syncthreads();
      }
    }

    // ---- Schur: S = A22 - L21 * U12 via 4x V_WMMA_F32_16X16X4_F32 ----
    {
      const int half = lane >> 4;
      const int q    = lane & 15;
      v8f acc = {};                      // accumulate L21*U12 with C starting at 0
      #pragma unroll
      for (int kk = 0; kk < NH; kk += 4) {
        const int ka = kk + 2 * half;    // ISA 7.12.2: lanes0-15 K={0,1}, lanes16-31 K={2,3}
        v2f a, bm;
        a.x  = As[(16 + q) * LDSP + ka];       // A-frag: L21, M=q (S row), K=ka..ka+1
        a.y  = As[(16 + q) * LDSP + ka + 1];
        bm.x = As[ka * LDSP + 16 + q];         // B-frag: U12, K, N=q (S col)
        bm.y = As[(ka + 1) * LDSP + 16 + q];
        acc = __builtin_amdgcn_wmma_f32_16x16x4_f32(
            /*neg_a=*/false, a, /*neg_b=*/false, bm,
            /*c_mod=*/(short)0, acc, /*reuse_a=*/false, /*reuse_b=*/false);
      }
      __syncthreads();
      // S = A22 - acc, written back into the A22 quadrant (C/D layout: VGPR r
      // holds row r + 8*half, lane holds column q)
      #pragma unroll
      for (int r = 0; r < 8; ++r) {
        const int m = r + 8 * half;
        float a22 = As[(16 + m) * LDSP + 16 + q];
        As[(16 + m) * LDSP + 16 + q] = a22 - acc[r];
      }
    }
    __syncthreads();

    // ---- unblocked pivoted LU on S (rows/cols 16..31) ----
    for (int k = 16; k < N; ++k) {
      float v = -1.0f; int p = 1 << 20;
      if (lane < 16) {
        int row = 16 + lane;
        if (row >= k) { v = fabsf(As[row * LDSP + k]); p = row; }
      }
      #pragma unroll
      for (int off = 16; off > 0; off >>= 1) {
        float ov = __shfl_xor(v, off);
        int   op = __shfl_xor(p, off);
        if (ov > v || (ov == v && op < p)) { v = ov; p = op; }
      }
      if (lane < 16) {                       // swap rows k<->p over cols 16..31
        int j = 16 + lane;
        float tk = As[k * LDSP + j];
        float tp = As[p * LDSP + j];
        As[k * LDSP + j] = tp;
        As[p * LDSP + j] = tk;
      }
      __syncthreads();
      float piv = As[k * LDSP + k];
      det *= piv;
      if (p != k) det = -det;
      float inv = (piv != 0.0f) ? 1.0f / piv : 0.0f;
      if (lane < 16) {
        int row = 16 + lane;
        if (row > k) {
          float lik = As[row * LDSP + k] * inv;
          for (int j = k + 1; j < N; ++j)
            As[row * LDSP + j] -= lik * As[k * LDSP + j];
        }
      }
      __syncthreads();
    }

    result *= det;
    __syncthreads();   // protect As before next spin's gather
  }

  if (lane == 0) out[bc] = result;
}

extern "C" void kernel_launch(void* const* d_in, const int* in_sizes, int n_in,
                              void* d_out, int out_size, void* d_ws, size_t ws_size,
                              hipStream_t stream) {
  (void)in_sizes; (void)n_in; (void)out_size; (void)d_ws; (void)ws_size;
  const float* mo     = (const float*)d_in[0];
  const int*   cfg_up = (const int*)d_in[1];   // jax int64 demotes to int32 (x64 disabled)
  const int*   cfg_dn = (const int*)d_in[2];
  float*       out    = (float*)d_out;

  dim3 grid(NB * NCONF);   // one wave32 workgroup per (batch, conf)
  dim3 block(32);
  hipLaunchKernelGGL(slater_det_kernel, grid, block, 0, stream,
                     mo, cfg_up, cfg_dn, out);
}